// GATConvMLP_26834955665496
// MI455X (gfx1250) — compile-verified
//
#include <hip/hip_runtime.h>
#include <cstddef>

#define NIMG 8192

typedef float v2f __attribute__((ext_vector_type(2)));
typedef float v8f __attribute__((ext_vector_type(8)));

// ---------------------------------------------------------------------------
// FP32 WMMA 16x16x4 helper (CDNA5 V_WMMA_F32_16X16X4_F32)
// ---------------------------------------------------------------------------
__device__ __forceinline__ v8f wmma_f32_16x16x4(v2f a, v2f b, v8f c) {
  // 8 args: (neg_a, A, neg_b, B, c_mod, C, reuse_a, reuse_b)
  return __builtin_amdgcn_wmma_f32_16x16x4_f32(false, a, false, b, (short)0, c,
                                               false, false);
}

// ---------------------------------------------------------------------------
// Conv1: x [N,3,32,32] * W1 [6,3,5,5] + b1, relu, maxpool2 -> h1 [N,6,14,14]
// ---------------------------------------------------------------------------
__global__ void conv1_relu_pool(const float* __restrict__ x,
                                const float* __restrict__ W1,
                                const float* __restrict__ b1,
                                float* __restrict__ h1) {
  __shared__ float sW[450];
  __shared__ float sB[6];
  for (int i = threadIdx.x; i < 450; i += blockDim.x) sW[i] = W1[i];
  if (threadIdx.x < 6) sB[threadIdx.x] = b1[threadIdx.x];
  __syncthreads();

  int idx = blockIdx.x * blockDim.x + threadIdx.x;
  const int total = NIMG * 6 * 14 * 14;
  if (idx >= total) return;
  int xo = idx % 14;
  int t  = idx / 14;
  int yo = t % 14; t /= 14;
  int c  = t % 6;
  int n  = t / 6;

  const float* xin = x + (size_t)n * 3 * 32 * 32;
  const float* w   = sW + c * 75;

  float mx = -3.4e38f;
  #pragma unroll
  for (int dy = 0; dy < 2; ++dy) {
    #pragma unroll
    for (int dx = 0; dx < 2; ++dx) {
      int iy = 2 * yo + dy;
      int ix = 2 * xo + dx;
      float s = sB[c];
      for (int ic = 0; ic < 3; ++ic) {
        #pragma unroll
        for (int ky = 0; ky < 5; ++ky) {
          const float* row = xin + ((size_t)ic * 32 + iy + ky) * 32 + ix;
          const float* wr  = w + ic * 25 + ky * 5;
          #pragma unroll
          for (int kx = 0; kx < 5; ++kx) s = fmaf(row[kx], wr[kx], s);
        }
      }
      mx = fmaxf(mx, s);
    }
  }
  h1[idx] = fmaxf(mx, 0.f);   // pool(relu(v)) == relu(pool(v))
}

// ---------------------------------------------------------------------------
// Conv2: h1 [N,6,14,14] * W2 [16,6,5,5] + b2, relu, maxpool2 -> h2 [N,16,5,5]
// ---------------------------------------------------------------------------
__global__ void conv2_relu_pool(const float* __restrict__ h1,
                                const float* __restrict__ W2,
                                const float* __restrict__ b2,
                                float* __restrict__ h2) {
  __shared__ float sW[2400];
  __shared__ float sB[16];
  for (int i = threadIdx.x; i < 2400; i += blockDim.x) sW[i] = W2[i];
  if (threadIdx.x < 16) sB[threadIdx.x] = b2[threadIdx.x];
  __syncthreads();

  int idx = blockIdx.x * blockDim.x + threadIdx.x;
  const int total = NIMG * 16 * 5 * 5;
  if (idx >= total) return;
  int xo = idx % 5;
  int t  = idx / 5;
  int yo = t % 5; t /= 5;
  int c  = t % 16;
  int n  = t / 16;

  const float* xin = h1 + (size_t)n * 6 * 14 * 14;
  const float* w   = sW + c * 150;

  float mx = -3.4e38f;
  #pragma unroll
  for (int dy = 0; dy < 2; ++dy) {
    #pragma unroll
    for (int dx = 0; dx < 2; ++dx) {
      int iy = 2 * yo + dy;
      int ix = 2 * xo + dx;
      float s = sB[c];
      for (int ic = 0; ic < 6; ++ic) {
        #pragma unroll
        for (int ky = 0; ky < 5; ++ky) {
          const float* row = xin + ((size_t)ic * 14 + iy + ky) * 14 + ix;
          const float* wr  = w + ic * 25 + ky * 5;
          #pragma unroll
          for (int kx = 0; kx < 5; ++kx) s = fmaf(row[kx], wr[kx], s);
        }
      }
      mx = fmaxf(mx, s);
    }
  }
  h2[idx] = fmaxf(mx, 0.f);
}

// ---------------------------------------------------------------------------
// Generic FC via fp32 WMMA: out[m,n] = act( sum_k A[m,k]*Wt[n,k] + bias[n] )
// One wave (32 threads) computes NT adjacent 16x16 output tiles, sharing the
// A fragment across tiles. Inner loop is branch-free: out-of-range N columns
// are clamped to row 0 of Wt and multiplied by a 0.0 mask, so EXEC stays
// all-1s (as WMMA requires) with no per-iteration saveexec.
// grid = (M/16, ceil(Nout/(16*NT))), block = 32.  K must be a multiple of 4.
// ---------------------------------------------------------------------------
template <int NT>
__global__ void fc_wmma(const float* __restrict__ A,
                        const float* __restrict__ Wt,
                        const float* __restrict__ bias,
                        float* __restrict__ out,
                        int K, int Nout, int do_relu) {
  const int m0   = blockIdx.x * 16;
  const int lane = threadIdx.x & 31;
  const int half = lane >> 4;   // 0: K sub-block {0,1}; 1: K sub-block {2,3}
  const int l15  = lane & 15;

  const float* arow = A + (size_t)(m0 + l15) * K + 2 * half;

  v8f          acc[NT];
  float        bmask[NT];
  int          nIdx[NT];
  const float* wrow[NT];
  #pragma unroll
  for (int t = 0; t < NT; ++t) {
    nIdx[t] = blockIdx.y * (16 * NT) + t * 16 + l15;
    const bool v = nIdx[t] < Nout;
    bmask[t] = v ? 1.f : 0.f;
    wrow[t]  = Wt + (size_t)(v ? nIdx[t] : 0) * K + 2 * half;
    acc[t]   = v8f{0.f, 0.f, 0.f, 0.f, 0.f, 0.f, 0.f, 0.f};
  }

  for (int k = 0; k < K; k += 4) {
    v2f a;
    a.x = arow[k];
    a.y = arow[k + 1];
    #pragma unroll
    for (int t = 0; t < NT; ++t) {
      v2f b;
      b.x = wrow[t][k] * bmask[t];
      b.y = wrow[t][k + 1] * bmask[t];
      acc[t] = wmma_f32_16x16x4(a, b, acc[t]);
    }
  }

  #pragma unroll
  for (int t = 0; t < NT; ++t) {
    if (nIdx[t] < Nout) {
      const float bv = bias ? bias[nIdx[t]] : 0.f;
      #pragma unroll
      for (int r = 0; r < 8; ++r) {
        float v = acc[t][r] + bv;
        if (do_relu) v = fmaxf(v, 0.f);
        out[(size_t)(m0 + r + 8 * half) * Nout + nIdx[t]] = v;
      }
    }
  }
}

// ---------------------------------------------------------------------------
// el/er: el[i] = z[i,:]@attn_l, er[i] = z[i,:]@attn_r
// ---------------------------------------------------------------------------
__global__ void attn_scores(const float* __restrict__ z,
                            const float* __restrict__ al,
                            const float* __restrict__ ar,
                            float* __restrict__ el,
                            float* __restrict__ er) {
  int i = blockIdx.x * blockDim.x + threadIdx.x;
  if (i >= NIMG) return;
  float sl = 0.f, sr = 0.f;
  #pragma unroll
  for (int c = 0; c < 10; ++c) {
    float v = z[(size_t)i * 10 + c];
    sl = fmaf(v, al[c], sl);
    sr = fmaf(v, ar[c], sr);
  }
  el[i] = sl;
  er[i] = sr;
}

// ---------------------------------------------------------------------------
// Single-block max reduction over el (for stable softmax).
// ---------------------------------------------------------------------------
__global__ void el_max(const float* __restrict__ el, float* __restrict__ elmax) {
  __shared__ float s[256];
  float m = -3.4e38f;
  for (int j = threadIdx.x; j < NIMG; j += 256) m = fmaxf(m, el[j]);
  s[threadIdx.x] = m;
  __syncthreads();
  for (int off = 128; off > 0; off >>= 1) {
    if (threadIdx.x < off)
      s[threadIdx.x] = fmaxf(s[threadIdx.x], s[threadIdx.x + off]);
    __syncthreads();
  }
  if (threadIdx.x == 0) elmax[0] = s[0];
}

// ---------------------------------------------------------------------------
// GAT aggregation. One block per destination node i.
//   e[i,j] = lrelu(er[i]+el[j], 0.2); row max = lrelu(er[i]+max_j el[j])
//   out[i] = (sum_j exp(e-m) * z[j]) / (sum_j exp(e-m)) + bg
// ---------------------------------------------------------------------------
__global__ void gat_aggregate(const float* __restrict__ z,
                              const float* __restrict__ el,
                              const float* __restrict__ er,
                              const float* __restrict__ elmax,
                              const float* __restrict__ bg,
                              float* __restrict__ out) {
  const int i = blockIdx.x;
  const float eri = er[i];
  float m = eri + elmax[0];
  m = (m > 0.f) ? m : 0.2f * m;   // leaky_relu, monotone -> this is the row max

  float wsum = 0.f;
  float acc[10];
  #pragma unroll
  for (int c = 0; c < 10; ++c) acc[c] = 0.f;

  for (int j = threadIdx.x; j < NIMG; j += blockDim.x) {
    float e = eri + el[j];
    e = (e > 0.f) ? e : 0.2f * e;
    float w = expf(e - m);
    wsum += w;
    const float* zj = z + (size_t)j * 10;
    #pragma unroll
    for (int c = 0; c < 10; ++c) acc[c] = fmaf(w, zj[c], acc[c]);
  }

  // wave32 intra-wave reduction
  #pragma unroll
  for (int off = 16; off > 0; off >>= 1) {
    wsum += __shfl_down(wsum, off);
    #pragma unroll
    for (int c = 0; c < 10; ++c) acc[c] += __shfl_down(acc[c], off);
  }

  __shared__ float sred[8][11];
  const int warp = threadIdx.x >> 5;
  const int ln   = threadIdx.x & 31;
  if (ln == 0) {
    sred[warp][0] = wsum;
    #pragma unroll
    for (int c = 0; c < 10; ++c) sred[warp][c + 1] = acc[c];
  }
  __syncthreads();
  if (threadIdx.x == 0) {
    float tw = 0.f;
    float ta[10];
    #pragma unroll
    for (int c = 0; c < 10; ++c) ta[c] = 0.f;
    for (int w = 0; w < 8; ++w) {
      tw += sred[w][0];
      #pragma unroll
      for (int c = 0; c < 10; ++c) ta[c] += sred[w][c + 1];
    }
    const float inv = 1.f / tw;
    #pragma unroll
    for (int c = 0; c < 10; ++c)
      out[(size_t)i * 10 + c] = ta[c] * inv + bg[c];
  }
}

// ---------------------------------------------------------------------------
// Orchestration
// ---------------------------------------------------------------------------
extern "C" void kernel_launch(void* const* d_in, const int* in_sizes, int n_in,
                              void* d_out, int out_size, void* d_ws, size_t ws_size,
                              hipStream_t stream) {
  const float* x      = (const float*)d_in[0];   // [8192,3,32,32]
  const float* W1     = (const float*)d_in[1];   // [6,3,5,5]
  const float* b1     = (const float*)d_in[2];   // [6]
  const float* W2     = (const float*)d_in[3];   // [16,6,5,5]
  const float* b2     = (const float*)d_in[4];   // [16]
  const float* Wf1    = (const float*)d_in[5];   // [120,400]
  const float* bf1    = (const float*)d_in[6];   // [120]
  const float* Wf2    = (const float*)d_in[7];   // [84,120]
  const float* bf2    = (const float*)d_in[8];   // [84]
  const float* Wg     = (const float*)d_in[9];   // [10,84]
  const float* attn_l = (const float*)d_in[10];  // [10]
  const float* attn_r = (const float*)d_in[11];  // [10]
  const float* bg     = (const float*)d_in[12];  // [10]
  float* out = (float*)d_out;                    // [8192,10]

  // Workspace layout (all 256B aligned)
  char* ws = (char*)d_ws;
  size_t ofs = 0;
  auto alloc = [&](size_t bytes) {
    char* p = ws + ofs;
    ofs += (bytes + 255) & ~(size_t)255;
    return (float*)p;
  };
  float* h1    = alloc((size_t)NIMG * 6 * 14 * 14 * 4);  // 38.5 MB
  float* h2    = alloc((size_t)NIMG * 400 * 4);          // 13.1 MB
  float* h3    = alloc((size_t)NIMG * 120 * 4);
  float* h4    = alloc((size_t)NIMG * 84 * 4);
  float* zbuf  = alloc((size_t)NIMG * 10 * 4);
  float* el    = alloc((size_t)NIMG * 4);
  float* er    = alloc((size_t)NIMG * 4);
  float* elmax = alloc(256);
  (void)ws_size; (void)in_sizes; (void)n_in; (void)out_size;

  // Conv stage 1
  {
    int total = NIMG * 6 * 14 * 14;
    conv1_relu_pool<<<(total + 255) / 256, 256, 0, stream>>>(x, W1, b1, h1);
  }
  // Conv stage 2 (output is flattened [N,400])
  {
    int total = NIMG * 16 * 5 * 5;
    conv2_relu_pool<<<(total + 255) / 256, 256, 0, stream>>>(h1, W2, b2, h2);
  }
  // FC1: [8192,400] x [400,120]^T + bf1, relu   (N padded 120 -> 128, 2 tiles/wave)
  fc_wmma<2><<<dim3(NIMG / 16, 4), 32, 0, stream>>>(h2, Wf1, bf1, h3, 400, 120, 1);
  // FC2: [8192,120] x [120,84]^T + bf2, relu    (N padded 84 -> 96, 2 tiles/wave)
  fc_wmma<2><<<dim3(NIMG / 16, 3), 32, 0, stream>>>(h3, Wf2, bf2, h4, 120, 84, 1);
  // z: [8192,84] x [84,10]^T                    (N padded 10 -> 16)
  fc_wmma<1><<<dim3(NIMG / 16, 1), 32, 0, stream>>>(h4, Wg, nullptr, zbuf, 84, 10, 0);
  // Attention scores
  attn_scores<<<(NIMG + 255) / 256, 256, 0, stream>>>(zbuf, attn_l, attn_r, el, er);
  // Global max of el for stable softmax
  el_max<<<1, 256, 0, stream>>>(el, elmax);
  // GAT aggregation
  gat_aggregate<<<NIMG, 256, 0, stream>>>(zbuf, el, er, elmax, bg, out);
}